// KernelTargetAlignmentLoss_39487929319478
// MI455X (gfx1250) — compile-verified
//
#include <hip/hip_runtime.h>
#include <hip/hip_bf16.h>

// ---------------------------------------------------------------------------
// KTA loss, N=8192 D=64, fused: never materializes the 8192x8192 K matrix.
// Gram matrix via V_WMMA_F32_16X16X4_F32 (full fp32 precision, matrix pipe).
// ---------------------------------------------------------------------------

typedef float v2f __attribute__((ext_vector_type(2)));
typedef float v8f __attribute__((ext_vector_type(8)));

#define KTA_N 8192
#define KTA_D 64
#define KTA_T 512                 // 8192/16 tiles per dimension
#define TILES_PER_BLOCK 8
#define NBLOCKS ((KTA_T * KTA_T) / TILES_PER_BLOCK)   // 32768

// ---------------------------------------------------------------------------
// Kernel 1: Xs = X * sqrt(params); sq[i] = sum_d Xs[i,d]^2
// 256 threads = 4 rows x 64 dims per block.
// ---------------------------------------------------------------------------
__global__ void kta_prep_kernel(const float* __restrict__ X,
                                const float* __restrict__ params,
                                float* __restrict__ Xs,
                                float* __restrict__ sq) {
    const int row = blockIdx.x * 4 + (threadIdx.x >> 6);
    const int d   = threadIdx.x & 63;
    const float v = X[row * KTA_D + d] * sqrtf(params[d]);
    Xs[row * KTA_D + d] = v;

    __shared__ float red[256];
    red[threadIdx.x] = v * v;
    __syncthreads();
#pragma unroll
    for (int s = 32; s >= 1; s >>= 1) {
        if (d < s) red[threadIdx.x] += red[threadIdx.x + s];
        __syncthreads();
    }
    if (d == 0) sq[row] = red[threadIdx.x];
}

// ---------------------------------------------------------------------------
// Kernel 2: one wave32 per 16x16 tile of K. Gram via fp32 WMMA, fused
// exp epilogue, deterministic per-block partial sums (no float atomics).
// Fragment layout per ISA 7.12.2 (32-bit A 16x4): lanes 0-15 carry M=0..15
// with VGPR pair = {K=0,K=1}; lanes 16-31 carry {K=2,K=3}. B mirrors A.
// C/D: vgpr r holds row m = r + 8*(lane/16), col n = lane%16.
// ---------------------------------------------------------------------------
__global__ void __launch_bounds__(256)
kta_tile_kernel(const float* __restrict__ Xs,
                const float* __restrict__ sq,
                const float* __restrict__ target,
                float* __restrict__ partials) {
    const int wave = threadIdx.x >> 5;               // 0..7
    const int lane = threadIdx.x & 31;
    const int tile = blockIdx.x * TILES_PER_BLOCK + wave;
    const int it   = tile >> 9;                      // / 512
    const int jt   = tile & (KTA_T - 1);             // % 512

    const int half = lane >> 4;                      // K-band select (0/1)
    const int l16  = lane & 15;
    const int rowA = (it << 4) + l16;                // A-matrix row (M)
    const int rowB = (jt << 4) + l16;                // B-matrix col (N)

    const float* pa = Xs + rowA * KTA_D + (half << 1);
    const float* pb = Xs + rowB * KTA_D + (half << 1);

    v8f acc = {0.f, 0.f, 0.f, 0.f, 0.f, 0.f, 0.f, 0.f};
#pragma unroll
    for (int t = 0; t < 16; ++t) {                   // K = 64 in steps of 4
        v2f a = *(const v2f*)(pa + 4 * t);
        v2f b = *(const v2f*)(pb + 4 * t);
        acc = __builtin_amdgcn_wmma_f32_16x16x4_f32(
            /*neg_a=*/false, a, /*neg_b=*/false, b,
            /*c_mod=*/(short)0, acc, /*reuse_a=*/false, /*reuse_b=*/false);
    }

    // Fused epilogue: d2 = max(sq_i + sq_j - 2*G, 0); K = exp(-d2); diag = 1.
    const int   gj  = (jt << 4) + l16;
    const float tj  = target[gj];
    const float sqj = sq[gj];
    float s1 = 0.f, s2 = 0.f;
#pragma unroll
    for (int r = 0; r < 8; ++r) {
        const int   gi  = (it << 4) + r + (half << 3);
        const float sqi = sq[gi];
        const float ti  = target[gi];
        float d2 = fmaxf(fmaf(-2.0f, acc[r], sqi + sqj), 0.0f);
        float Kv = __expf(-d2);
        if (gi == gj) Kv = 1.0f;                     // reference forces diagonal
        s1 = fmaf(Kv, ti * tj, s1);
        s2 = fmaf(Kv, Kv, s2);
    }

    // wave32 butterfly reduction
#pragma unroll
    for (int off = 16; off >= 1; off >>= 1) {
        s1 += __shfl_xor(s1, off, 32);
        s2 += __shfl_xor(s2, off, 32);
    }

    __shared__ float r1[TILES_PER_BLOCK];
    __shared__ float r2[TILES_PER_BLOCK];
    if (lane == 0) { r1[wave] = s1; r2[wave] = s2; }
    __syncthreads();
    if (threadIdx.x == 0) {
        float t1 = 0.f, t2 = 0.f;
#pragma unroll
        for (int w = 0; w < TILES_PER_BLOCK; ++w) { t1 += r1[w]; t2 += r2[w]; }
        partials[blockIdx.x]           = t1;         // sum K * t_i * t_j
        partials[NBLOCKS + blockIdx.x] = t2;         // sum K * K
    }
}

// ---------------------------------------------------------------------------
// Kernel 3: final reduction over per-block partials; write -KTA scalar.
// ---------------------------------------------------------------------------
__global__ void kta_finalize_kernel(const float* __restrict__ partials,
                                    float* __restrict__ out) {
    __shared__ float red1[256];
    __shared__ float red2[256];
    float s1 = 0.f, s2 = 0.f;
    for (int i = threadIdx.x; i < NBLOCKS; i += 256) {
        s1 += partials[i];
        s2 += partials[NBLOCKS + i];
    }
    red1[threadIdx.x] = s1;
    red2[threadIdx.x] = s2;
    __syncthreads();
#pragma unroll
    for (int s = 128; s >= 1; s >>= 1) {
        if (threadIdx.x < s) {
            red1[threadIdx.x] += red1[threadIdx.x + s];
            red2[threadIdx.x] += red2[threadIdx.x + s];
        }
        __syncthreads();
    }
    if (threadIdx.x == 0) {
        out[0] = -(red1[0] / ((float)KTA_N * sqrtf(red2[0])));
    }
}

// ---------------------------------------------------------------------------
extern "C" void kernel_launch(void* const* d_in, const int* in_sizes, int n_in,
                              void* d_out, int out_size, void* d_ws, size_t ws_size,
                              hipStream_t stream) {
    const float* X      = (const float*)d_in[0];   // [8192, 64]
    const float* target = (const float*)d_in[1];   // [8192]
    const float* params = (const float*)d_in[2];   // [64]
    float* out = (float*)d_out;                    // [1]

    // Workspace layout (all fp32):
    //   Xs       : N*D        = 524288 floats (2 MB)
    //   sq       : N          =   8192 floats
    //   partials : 2*NBLOCKS  =  65536 floats
    float* Xs       = (float*)d_ws;
    float* sq       = Xs + KTA_N * KTA_D;
    float* partials = sq + KTA_N;

    kta_prep_kernel<<<KTA_N / 4, 256, 0, stream>>>(X, params, Xs, sq);
    kta_tile_kernel<<<NBLOCKS, 256, 0, stream>>>(Xs, sq, target, partials);
    kta_finalize_kernel<<<1, 256, 0, stream>>>(partials, out);
}